// HiPPOLayer_28097676050640
// MI455X (gfx1250) — compile-verified
//
#include <hip/hip_runtime.h>
#include <stdint.h>

// Problem dims (fixed by reference)
#define BATCH 32
#define SEQ   512
#define FEAT  256
#define NST   8
#define NCH   8     // chunks per sequence
#define CLEN  64    // SEQ / NCH

typedef __attribute__((ext_vector_type(2))) float v2f;
typedef __attribute__((ext_vector_type(8))) float v8f;

// ---------------------------------------------------------------------------
// K1: build WT16[i][j] = (B_d @ A_d^(63-i))[j] for j<8, 0 for j>=8 (i=0..63),
// and M64 = A_d^64.  Log-doubling: w_{L+j} = w_j @ A^L ; A^{2L} = A^L @ A^L.
// WT16 is pre-padded to 16 columns so K2's B-operand loads are branch-free.
// ---------------------------------------------------------------------------
__global__ void k1_tables(const float* __restrict__ A, const float* __restrict__ Bv,
                          float* __restrict__ WT16, float* __restrict__ M64) {
  __shared__ float W[64][8];
  __shared__ float Mc[8][8];
  __shared__ float Mt[8][8];
  int t = threadIdx.x;
  if (t < 64) Mc[t >> 3][t & 7] = A[t];
  if (t < 8)  W[0][t] = Bv[t];          // w_0 = B
  __syncthreads();
  int L = 1;
  for (int s = 0; s < 6; ++s) {
    if (t < L * 8) {                    // rows L..2L-1
      int j = t >> 3, m = t & 7;
      float acc = 0.f;
#pragma unroll
      for (int n = 0; n < 8; ++n) acc += W[j][n] * Mc[n][m];
      W[L + j][m] = acc;
    }
    if (t < 64) {                       // Mt = Mc @ Mc
      int n = t >> 3, m = t & 7;
      float acc = 0.f;
#pragma unroll
      for (int k = 0; k < 8; ++k) acc += Mc[n][k] * Mc[k][m];
      Mt[n][m] = acc;
    }
    __syncthreads();
    if (t < 64) Mc[t >> 3][t & 7] = Mt[t >> 3][t & 7];
    __syncthreads();
    L <<= 1;
  }
  if (t < 64) M64[t] = Mc[t >> 3][t & 7];      // A^64
  for (int i = t; i < 1024; i += 256) {        // padded + reversed
    int row = i >> 4, col = i & 15;
    WT16[i] = (col < 8) ? W[63 - row][col] : 0.f;
  }
}

// ---------------------------------------------------------------------------
// K2: chunk injection via WMMA f32 16x16x4.
//   d(16f x 8m) = X(16f x 64t) @ W(64t x 8m)       per (b, p, f-tile)
// A-operand (16x4 f32): lane = M (=f), vgpr0 holds K=2*half, vgpr1 K=2*half+1
// B-operand (4x16 f32): lane = N (=m), vgpr0 holds K=2*half, vgpr1 K=2*half+1
// D layout: vgpr v -> M = v + 8*half, N = lane&15
// ---------------------------------------------------------------------------
__global__ void k2_inject(const float* __restrict__ x, const float* __restrict__ WT16,
                          float* __restrict__ dtab) {
  int wave = threadIdx.x >> 5;
  int lane = threadIdx.x & 31;
  int tile = blockIdx.x * 8 + wave;     // 0..4095 = b*128 + p*16 + ftile
  int ftile = tile & 15;
  int p = (tile >> 4) & 7;
  int b = tile >> 7;
  int half = lane >> 4;
  int lr = lane & 15;

  const float* xb = x + ((size_t)b * SEQ + (size_t)p * CLEN) * FEAT + ftile * 16 + lr;
  const float* wb = WT16 + 2 * half * 16 + lr;   // row k0, col lr (pre-padded)
  v8f acc = {0.f, 0.f, 0.f, 0.f, 0.f, 0.f, 0.f, 0.f};
#pragma unroll
  for (int kk = 0; kk < 16; ++kk) {
    int k0 = kk * 4 + 2 * half;
    v2f a, bb;
    a.x = xb[(size_t)k0 * FEAT];
    a.y = xb[(size_t)(k0 + 1) * FEAT];
    bb.x = wb[(kk * 4) * 16];
    bb.y = wb[(kk * 4) * 16 + 16];
    acc = __builtin_amdgcn_wmma_f32_16x16x4_f32(false, a, false, bb,
                                                (short)0, acc, false, false);
  }
  if (lr < 8) {  // N = state index m = lr; M = v + 8*half -> feature row
    float* dp = dtab + (((size_t)(b * NCH + p) * FEAT) + ftile * 16 + 8 * half) * NST + lr;
#pragma unroll
    for (int v = 0; v < 8; ++v) dp[(size_t)v * NST] = acc[v];
  }
}

// ---------------------------------------------------------------------------
// K3: boundary scan.  Stab[b,p,f,:] = state BEFORE chunk p.
//   s_{p+1} = s_p @ M64 + d[b,p,f,:]
// ---------------------------------------------------------------------------
__global__ void k3_scan(const float* __restrict__ dtab, const float* __restrict__ M64,
                        float* __restrict__ Stab) {
  int f = threadIdx.x;
  int b = blockIdx.x;
  float M[64];
#pragma unroll
  for (int i = 0; i < 64; ++i) M[i] = M64[i];   // uniform -> s_loads
  float s[8] = {0, 0, 0, 0, 0, 0, 0, 0};
  for (int p = 0; p < NCH; ++p) {
    size_t base = (((size_t)(b * NCH + p) * FEAT) + f) * NST;
#pragma unroll
    for (int m = 0; m < 8; ++m) Stab[base + m] = s[m];
    float sn[8];
#pragma unroll
    for (int m = 0; m < 8; ++m) {
      float acc = dtab[base + m];
#pragma unroll
      for (int n = 0; n < 8; ++n) acc += s[n] * M[n * 8 + m];
      sn[m] = acc;
    }
#pragma unroll
    for (int m = 0; m < 8; ++m) s[m] = sn[m];
  }
}

// ---------------------------------------------------------------------------
// K4: main recurrence, one (b, chunk) per block; 2048 waves total.
// x staged into LDS with double-buffered async-to-LDS (per-wave self staging,
// so only s_wait_asynccnt is needed, no barriers).
// ---------------------------------------------------------------------------
__device__ __forceinline__ void stage_group(const float* gbase, unsigned ldsbase,
                                            unsigned fbyte) {
#pragma unroll
  for (int tt = 0; tt < 16; ++tt) {
    unsigned ldsoff = ldsbase + (unsigned)(tt * FEAT * 4);
    unsigned long long sb = (unsigned long long)(uintptr_t)(gbase + (size_t)tt * FEAT);
    asm volatile("global_load_async_to_lds_b32 %0, %1, %2"
                 :: "v"(ldsoff), "v"(fbyte), "s"(sb) : "memory");
  }
}

__global__ void k4_main(const float* __restrict__ x, const float* __restrict__ A,
                        const float* __restrict__ Bv, const float* __restrict__ Stab,
                        float* __restrict__ out) {
  __shared__ float xbuf[2][16][FEAT];
  int f = threadIdx.x;
  int p = blockIdx.x & 7;
  int b = blockIdx.x >> 3;

  float Areg[64];
#pragma unroll
  for (int i = 0; i < 64; ++i) Areg[i] = A[i];  // uniform -> SGPRs
  float Breg[8];
#pragma unroll
  for (int m = 0; m < 8; ++m) Breg[m] = Bv[m];

  float c[8];
  {
    size_t sbase = (((size_t)(b * NCH + p) * FEAT) + f) * NST;
#pragma unroll
    for (int m = 0; m < 8; ++m) c[m] = Stab[sbase + m];
  }

  const float* xsrc = x + ((size_t)b * SEQ + (size_t)p * CLEN) * FEAT;
  float* osrc = out + (((size_t)b * SEQ + (size_t)p * CLEN) * FEAT + f) * NST;

  unsigned l0 = (unsigned)(uintptr_t)(&xbuf[0][0][0]) + (unsigned)(f * 4);
  unsigned l1 = l0 + (unsigned)(16 * FEAT * 4);
  unsigned fbyte = (unsigned)(f * 4);

  stage_group(xsrc, l0, fbyte);                 // prologue: group 0 -> buf0
  for (int g = 0; g < 4; ++g) {
    if (g < 3) {
      stage_group(xsrc + (size_t)(g + 1) * 16 * FEAT, ((g + 1) & 1) ? l1 : l0, fbyte);
      asm volatile("s_wait_asynccnt 0x10" ::: "memory");  // group g complete
    } else {
      asm volatile("s_wait_asynccnt 0x0" ::: "memory");
    }
    const float* xl = &xbuf[g & 1][0][f];
    float* op = osrc + (size_t)g * 16 * FEAT * NST;
#pragma unroll 4
    for (int tt = 0; tt < 16; ++tt) {
      float xv = xl[(size_t)tt * FEAT];
      float cn[8];
#pragma unroll
      for (int m = 0; m < 8; ++m) {
        float acc = Breg[m] * xv;
#pragma unroll
        for (int n = 0; n < 8; ++n) acc += c[n] * Areg[n * 8 + m];
        cn[m] = acc;
      }
#pragma unroll
      for (int m = 0; m < 8; ++m) c[m] = cn[m];
      float4 lo = make_float4(c[0], c[1], c[2], c[3]);
      float4 hi = make_float4(c[4], c[5], c[6], c[7]);
      *(float4*)(op + (size_t)tt * FEAT * NST)     = lo;
      *(float4*)(op + (size_t)tt * FEAT * NST + 4) = hi;
    }
  }
}

// ---------------------------------------------------------------------------
extern "C" void kernel_launch(void* const* d_in, const int* in_sizes, int n_in,
                              void* d_out, int out_size, void* d_ws, size_t ws_size,
                              hipStream_t stream) {
  const float* x  = (const float*)d_in[0];   // (32,512,256)
  const float* Ad = (const float*)d_in[1];   // (8,8)
  const float* Bd = (const float*)d_in[2];   // (8,)
  float* out = (float*)d_out;                // (32,512,2048)

  float* ws    = (float*)d_ws;
  float* WT16  = ws;                         // 64*16 = 1024 floats (zero-padded)
  float* M64   = ws + 1024;                  // 64 floats
  float* dtab  = ws + 1088;                  // 32*8*256*8 = 524288 floats
  float* Stab  = dtab + (size_t)BATCH * NCH * FEAT * NST;  // 524288 floats

  k1_tables<<<1, 256, 0, stream>>>(Ad, Bd, WT16, M64);
  k2_inject<<<512, 256, 0, stream>>>(x, WT16, dtab);
  k3_scan<<<BATCH, 256, 0, stream>>>(dtab, M64, Stab);
  k4_main<<<BATCH * NCH, 256, 0, stream>>>(x, Ad, Bd, Stab, out);
}